// Sbert_27797028340322
// MI455X (gfx1250) — compile-verified
//
#include <hip/hip_runtime.h>

typedef __attribute__((ext_vector_type(2))) float v2f;
typedef __attribute__((ext_vector_type(8))) float v8f;

#define BATCH 4096
#define SEQL  100
#define DIM   768
#define DGRPS 6          // 6 blocks per batch, each covers 128 D-columns

// One wave computes out[b, d0 .. d0+15] as an fp32 WMMA chain over L.
// A (16x4)  : w[l0..l0+3] broadcast down all 16 M rows
// B (4x16)  : hidden[b, l0..l0+3, d0..d0+15]
// C (16x16) : running fp32 accumulator; every M row identical by construction.
__global__ __launch_bounds__(256)
void idf_pool_wmma_f32(const float* __restrict__ hidden,
                       const int*   __restrict__ ids,
                       const int*   __restrict__ mask,
                       const float* __restrict__ idf,
                       float*       __restrict__ out)
{
    __shared__ float s_w[SEQL];
    __shared__ float s_msum;

    const int b    = blockIdx.x / DGRPS;
    const int dgrp = blockIdx.x % DGRPS;
    const int tid  = threadIdx.x;

    if (tid == 0) s_msum = 0.0f;
    __syncthreads();

    // Per-batch token weights: w[l] = idf[ids[b,l]] * mask[b,l]; also sum the mask.
    if (tid < SEQL) {
        const int   id = ids[b * SEQL + tid];
        const float m  = (float)mask[b * SEQL + tid];
        s_w[tid] = idf[id] * m;
        atomicAdd(&s_msum, m);             // ds_add_f32
    }
    __syncthreads();

    const float inv = 1.0f / fmaxf(s_msum, 1e-9f);

    const int wave = tid >> 5;             // 0..7  -> which 16-wide D tile
    const int lane = tid & 31;
    const int half = lane >> 4;            // half-wave selects K pair {0,1} vs {2,3}
    const int ln   = lane & 15;            // N index within the 16x16 tile

    const int d0 = dgrp * 128 + wave * 16;
    const float* __restrict__ hb = hidden + (size_t)b * (SEQL * DIM);

    v8f c = {};                            // fp32 accumulator tile

    #pragma unroll
    for (int l0 = 0; l0 < SEQL; l0 += 4) {
        const int lA = l0 + half * 2;      // K row held by this half-wave (VGPR0)

        // A operand: lanes 0-15 get {w[l0+0], w[l0+1]}, lanes 16-31 {w[l0+2], w[l0+3]}
        v2f a;
        a.x = s_w[lA];
        a.y = s_w[lA + 1];

        // B operand: matching hidden rows, 16 contiguous floats per half-wave row.
        // Streamed once -> non-temporal.
        const float* p = hb + (size_t)lA * DIM + d0 + ln;
        v2f bm;
        bm.x = __builtin_nontemporal_load(p);
        bm.y = __builtin_nontemporal_load(p + DIM);

        // D = A x B + C  (exact fp32, matches reference einsum semantics)
        c = __builtin_amdgcn_wmma_f32_16x16x4_f32(
                /*neg_a=*/false, a, /*neg_b=*/false, bm,
                /*c_mod=*/(short)0, c, /*reuse_a=*/false, /*reuse_b=*/false);
    }

    // C/D layout: VGPR0, lanes 0-15 = (M=0, N=lane); all M rows identical.
    if (lane < 16) {
        out[(size_t)b * DIM + d0 + ln] = c[0] * inv;
    }
}

extern "C" void kernel_launch(void* const* d_in, const int* in_sizes, int n_in,
                              void* d_out, int out_size, void* d_ws, size_t ws_size,
                              hipStream_t stream) {
    const float* hidden = (const float*)d_in[0];   // [B, L, D] fp32
    const int*   ids    = (const int*)  d_in[1];   // [B, L] int32
    const int*   mask   = (const int*)  d_in[2];   // [B, L] int32
    const float* idf    = (const float*)d_in[3];   // [V] fp32
    float*       out    = (float*)d_out;           // [B, D] fp32

    const int grid = BATCH * DGRPS;                // 24576 blocks
    idf_pool_wmma_f32<<<grid, 256, 0, stream>>>(hidden, ids, mask, idf, out);
}